// EntanglementTransform_62998580297889
// MI455X (gfx1250) — compile-verified
//
#include <hip/hip_runtime.h>

typedef __attribute__((ext_vector_type(2))) float v2f;
typedef __attribute__((ext_vector_type(8))) float v8f;

#define QN 64
#define HN 2048
#define EPSF 1e-12f

// ---------------------------------------------------------------------------
// Phase 1: factor[q,h] = prod_{j>q} W[q,j,h] * prod_{i<q} W[i,q,h]
// One thread per (q,h). Consecutive threads -> consecutive h: fully coalesced.
// Total traffic ~33 MB, ~1.5 us -- negligible vs phase 2.
// ---------------------------------------------------------------------------
__global__ void __launch_bounds__(256)
factor_kernel(const float* __restrict__ W, float* __restrict__ factor) {
    const int idx = blockIdx.x * 256 + threadIdx.x;   // 0 .. QN*HN-1
    const int h = idx & (HN - 1);
    const int k = idx >> 11;                          // h-dim is 2048
    float f = 1.0f;
    // j > k terms: W[k, j, h]
    for (int j = k + 1; j < QN; ++j)
        f *= W[((size_t)(k * QN + j)) * HN + h];
    // i < k terms: W[i, k, h]
    for (int i = 0; i < k; ++i)
        f *= W[((size_t)(i * QN + k)) * HN + h];
    factor[idx] = f;
}

// ---------------------------------------------------------------------------
// Phase 2: y = x * factor ; out = y / max(||y||_2 (over H), eps)
// One 256-thread block (8 wave32) per (batch,q) row of 2048 floats.
// Each thread keeps its 8 y-values in registers (single pass over x).
// Intra-wave sum-of-squares reduction: two chained V_WMMA_F32_16X16X4_F32
// against an all-ones B. After WMMA #1, lane L holds t[0..7] (L<16) or
// t[8..15] (L>=16) where t[m] = s[m]+s[m+16]; summing the 8 accumulator
// regs and running WMMA #2 broadcasts the full 32-lane sum to every lane.
// ---------------------------------------------------------------------------
__global__ void __launch_bounds__(256)
normalize_kernel(const float* __restrict__ x,
                 const float* __restrict__ factor,
                 float* __restrict__ out) {
    const int row = blockIdx.x;            // b*QN + q
    const int q   = row & (QN - 1);
    const int t   = threadIdx.x;           // 0..255
    const size_t xbase = (size_t)row * HN + (size_t)t * 8;
    const size_t fbase = (size_t)q   * HN + (size_t)t * 8;

    // two B128 loads from x, two from factor (factor row is L2-resident)
    const float4 xa = *(const float4*)(x + xbase);
    const float4 xb = *(const float4*)(x + xbase + 4);
    const float4 fa = *(const float4*)(factor + fbase);
    const float4 fb = *(const float4*)(factor + fbase + 4);

    float4 ya, yb;
    ya.x = xa.x * fa.x;  ya.y = xa.y * fa.y;
    ya.z = xa.z * fa.z;  ya.w = xa.w * fa.w;
    yb.x = xb.x * fb.x;  yb.y = xb.y * fb.y;
    yb.z = xb.z * fb.z;  yb.w = xb.w * fb.w;

    float s = ya.x * ya.x + ya.y * ya.y + ya.z * ya.z + ya.w * ya.w
            + yb.x * yb.x + yb.y * yb.y + yb.z * yb.z + yb.w * yb.w;

    // ---- wave32 reduction via 2x WMMA (B = ones, EXEC all-1s here) ----
    v2f bones; bones.x = 1.0f; bones.y = 1.0f;

    v2f a1; a1.x = s; a1.y = 0.0f;
    v8f c1 = {};
    c1 = __builtin_amdgcn_wmma_f32_16x16x4_f32(
        false, a1, false, bones, (short)0, c1, false, false);
    float u = c1[0] + c1[1] + c1[2] + c1[3] + c1[4] + c1[5] + c1[6] + c1[7];

    v2f a2; a2.x = u; a2.y = 0.0f;
    v8f c2 = {};
    c2 = __builtin_amdgcn_wmma_f32_16x16x4_f32(
        false, a2, false, bones, (short)0, c2, false, false);
    const float wave_sum = c2[0];          // full 32-lane sum, in every lane

    // ---- combine the 8 waves through LDS ----
    __shared__ float lds[8];
    const int wave = t >> 5;
    if ((t & 31) == 0) lds[wave] = wave_sum;
    __syncthreads();
    float total = 0.0f;
#pragma unroll
    for (int w = 0; w < 8; ++w) total += lds[w];

    const float rnorm = 1.0f / fmaxf(sqrtf(total), EPSF);

    ya.x *= rnorm; ya.y *= rnorm; ya.z *= rnorm; ya.w *= rnorm;
    yb.x *= rnorm; yb.y *= rnorm; yb.z *= rnorm; yb.w *= rnorm;
    *(float4*)(out + xbase)     = ya;
    *(float4*)(out + xbase + 4) = yb;
}

extern "C" void kernel_launch(void* const* d_in, const int* in_sizes, int n_in,
                              void* d_out, int out_size, void* d_ws, size_t ws_size,
                              hipStream_t stream) {
    const float* x = (const float*)d_in[0];                 // [B, 64, 2048]
    const float* W = (const float*)d_in[1];                 // [64, 64, 2048]
    float* out     = (float*)d_out;                         // [B, 64, 2048]
    float* factor  = (float*)d_ws;                          // [64, 2048] scratch

    const int batch = in_sizes[0] / (QN * HN);

    factor_kernel<<<(QN * HN) / 256, 256, 0, stream>>>(W, factor);
    normalize_kernel<<<batch * QN, 256, 0, stream>>>(x, factor, out);
}